// MagnusLadder_51857435132104
// MI455X (gfx1250) — compile-verified
//
#include <hip/hip_runtime.h>

// MagnusLadder: x_{t+1} = expm(A*dt) @ x_t, emit all T states.
// Time-parallel: Q_j = E^(2^j); block g builds checkpoint E^(32g) x0 from the
// bits of g, then runs 31 serial WMMA_F32_16X16X4_F32 steps.
// v2: kt-outer / nt-inner WMMA ordering -> two independent accumulator
// chains (2x ILP on the serial critical path) and single A-fragment load
// per k-tile in the checkpoint matmuls.

typedef float v2f __attribute__((ext_vector_type(2)));
typedef float v8f __attribute__((ext_vector_type(8)));

#define DIM      256
#define NB       32          // batch columns
#define TSTEPS   8192
#define CH       32          // steps per chunk
#define NG       (TSTEPS/CH) // 256 blocks
#define RS       260         // padded LDS row stride (floats) for xT[32][RS]
#define NTHREADS 512         // 16 waves (wave32)
#define MATSZ    (DIM*DIM)

// ws layout in floats
#define WS_ADT 0
#define WS_P0  (1*MATSZ)
#define WS_P1  (2*MATSZ)
#define WS_Q0  (3*MATSZ)     // Q[j] = E^(2^j), j = 0..12 ; Q[0] = E

// ---------------------------------------------------------------- init -----
__global__ void ml_init(const float* __restrict__ t,
                        const float* __restrict__ A,
                        float* __restrict__ ws) {
  const int i = blockIdx.x;        // row
  const int j = threadIdx.x;       // col
  const float dt = t[1] - t[0];
  const float a  = A[i * DIM + j] * dt;
  ws[WS_ADT + i * DIM + j] = a;                              // X = A*dt
  ws[WS_P0  + i * DIM + j] = a;                              // P1 = X
  ws[WS_Q0  + i * DIM + j] = a + (i == j ? 1.0f : 0.0f);     // E  = I + X
}

// ------------------------------------------- plain 256x256x256 matmul -----
// Z = (X @ Y) * scale ; optionally Eacc += Z   (Taylor terms / squarings)
__global__ void ml_matmul(const float* X, const float* Y,
                          float* __restrict__ Z, float* Eacc, float scale) {
  const int i = blockIdx.x;
  const int j = threadIdx.x;
  float s = 0.0f;
#pragma unroll 8
  for (int k = 0; k < DIM; ++k)
    s = fmaf(X[i * DIM + k], Y[k * DIM + j], s);
  s *= scale;
  Z[i * DIM + j] = s;
  if (Eacc) Eacc[i * DIM + j] += s;
}

// ---------------------------------------------------------------- main -----
__launch_bounds__(NTHREADS, 1)
__global__ void ml_main(const float* __restrict__ x0,
                        const float* __restrict__ ws,
                        float* __restrict__ out) {
  // x state, transposed xT[b][d], padded rows (stride-4 banks -> conflict-free
  // b64 B-fragment loads), double buffered.
  __shared__ float xs[2][NB * RS];

  const int tid  = threadIdx.x;
  const int lane = tid & 31;
  const int w    = tid >> 5;     // wave id 0..15 -> output rows 16w..16w+15
  const int rlo  = lane & 15;
  const int hi   = lane >> 4;    // half-wave selects K-pair {0,1} vs {2,3}
  const int m0   = w * 16;
  const int g    = blockIdx.x;
  const long t0  = (long)g * CH;

  // stage x0 into xs[0]: xs[0][b*RS + d] = x0[d*NB + b]
  for (int e = 0; e < (DIM * NB) / NTHREADS; ++e) {
    const int idx = tid + e * NTHREADS;      // = d*NB + b
    xs[0][(idx & 31) * RS + (idx >> 5)] = x0[idx];
  }
  __syncthreads();

  int cur = 0;

  // ---- checkpoint: x = (prod over set bits i of g) Q_{5+i} @ x -----------
  for (int bit = 0; bit < 8; ++bit) {
    if ((g >> bit) & 1) {                    // uniform branch: EXEC stays full
      const float* Q  = ws + WS_Q0 + (5 + bit) * MATSZ;
      const float* xc = xs[cur];
      float*       xn = xs[cur ^ 1];
      v8f acc0 = {0.f, 0.f, 0.f, 0.f, 0.f, 0.f, 0.f, 0.f};
      v8f acc1 = acc0;
#pragma unroll
      for (int kt = 0; kt < 64; ++kt) {
        v2f a  = *(const v2f*)&Q[(m0 + rlo) * DIM + kt * 4 + 2 * hi];
        v2f b0 = *(const v2f*)&xc[(rlo)      * RS + kt * 4 + 2 * hi];
        v2f b1 = *(const v2f*)&xc[(16 + rlo) * RS + kt * 4 + 2 * hi];
        acc0 = __builtin_amdgcn_wmma_f32_16x16x4_f32(
            false, a, false, b0, (short)0, acc0, false, false);
        acc1 = __builtin_amdgcn_wmma_f32_16x16x4_f32(
            false, a, false, b1, (short)0, acc1, false, false);
      }
#pragma unroll
      for (int r = 0; r < 8; ++r) {
        xn[(rlo)      * RS + m0 + r + 8 * hi] = acc0[r];
        xn[(16 + rlo) * RS + m0 + r + 8 * hi] = acc1[r];
      }
      __syncthreads();
      cur ^= 1;
    }
  }

  // ---- emit checkpoint at t = t0 ----------------------------------------
  {
    const float* xc = xs[cur];
    for (int e = 0; e < (DIM * NB) / NTHREADS; ++e) {
      const int idx = tid + e * NTHREADS;    // = d*NB + b
      out[t0 * (long)(DIM * NB) + idx] = xc[(idx & 31) * RS + (idx >> 5)];
    }
  }

  // ---- load E = Q[0] A-fragments into registers (64 x v2f = 128 VGPRs) ---
  const float* E = ws + WS_Q0;
  v2f ef[64];
#pragma unroll
  for (int kt = 0; kt < 64; ++kt)
    ef[kt] = *(const v2f*)&E[(m0 + rlo) * DIM + kt * 4 + 2 * hi];

  // ---- 31 serial steps: x <- E @ x, write each state ---------------------
  for (int s = 1; s < CH; ++s) {
    const float* xc = xs[cur];
    float*       xn = xs[cur ^ 1];
    float*       og = out + (t0 + s) * (long)(DIM * NB);
    v8f acc0 = {0.f, 0.f, 0.f, 0.f, 0.f, 0.f, 0.f, 0.f};
    v8f acc1 = acc0;
#pragma unroll
    for (int kt = 0; kt < 64; ++kt) {
      v2f b0 = *(const v2f*)&xc[(rlo)      * RS + kt * 4 + 2 * hi];
      v2f b1 = *(const v2f*)&xc[(16 + rlo) * RS + kt * 4 + 2 * hi];
      acc0 = __builtin_amdgcn_wmma_f32_16x16x4_f32(
          false, ef[kt], false, b0, (short)0, acc0, false, false);
      acc1 = __builtin_amdgcn_wmma_f32_16x16x4_f32(
          false, ef[kt], false, b1, (short)0, acc1, false, false);
    }
#pragma unroll
    for (int r = 0; r < 8; ++r) {
      const int d = m0 + r + 8 * hi;
      xn[(rlo)      * RS + d] = acc0[r];
      xn[(16 + rlo) * RS + d] = acc1[r];
      og[d * NB + rlo]        = acc0[r];
      og[d * NB + 16 + rlo]   = acc1[r];
    }
    __syncthreads();
    cur ^= 1;
  }
}

// -------------------------------------------------------------- launch -----
extern "C" void kernel_launch(void* const* d_in, const int* in_sizes, int n_in,
                              void* d_out, int out_size, void* d_ws, size_t ws_size,
                              hipStream_t stream) {
  const float* t  = (const float*)d_in[0];
  const float* x0 = (const float*)d_in[1];
  const float* A  = (const float*)d_in[2];
  float* out = (float*)d_out;
  float* ws  = (float*)d_ws;   // needs 16 * 256KB = 4 MB

  // E = I + X + X^2/2! + ... + X^6/6!   (||X||~0.011 -> trunc err ~1e-18)
  ml_init<<<DIM, DIM, 0, stream>>>(t, A, ws);
  float* P[2] = { ws + WS_P0, ws + WS_P1 };
  int cur = 0;
  for (int k = 2; k <= 6; ++k) {
    ml_matmul<<<DIM, DIM, 0, stream>>>(P[cur], ws + WS_ADT, P[cur ^ 1],
                                       ws + WS_Q0, 1.0f / (float)k);
    cur ^= 1;
  }
  // Q[j] = Q[j-1]^2, j = 1..12
  for (int j = 1; j <= 12; ++j) {
    ml_matmul<<<DIM, DIM, 0, stream>>>(ws + WS_Q0 + (j - 1) * MATSZ,
                                       ws + WS_Q0 + (j - 1) * MATSZ,
                                       ws + WS_Q0 + j * MATSZ,
                                       (float*)nullptr, 1.0f);
  }
  // 256 time-chunks of 32 steps each
  ml_main<<<NG, NTHREADS, 0, stream>>>(x0, ws, out);
}